// N_REN_8564164788789
// MI455X (gfx1250) — compile-verified
//
#include <hip/hip_runtime.h>

typedef __attribute__((ext_vector_type(2))) float v2f;
typedef __attribute__((ext_vector_type(8))) float v8f;

#define WMMA_F32(a, b, c) \
  __builtin_amdgcn_wmma_f32_16x16x4_f32(false, (a), false, (b), (short)0, (c), false, false)

static constexpr int kNX = 256;
static constexpr int kNQ = 64;
static constexpr int kNH = 320;  // NX + NQ

// ---------------------------------------------------------------------------
// gfx1250 async global->LDS path (ASYNCcnt-tracked). The clang builtin wants
// CUDA-LangAS-qualified int4 pointers that HIP source cannot spell, so we use
// inline asm (ISA 15.18.3 opcode 98: GLOBAL_LOAD_ASYNC_TO_LDS_B128).
// Gated on the toolchain declaring the gfx1250 async builtins.
// ---------------------------------------------------------------------------
#if __has_builtin(__builtin_amdgcn_global_load_async_to_lds_b128) && \
    __has_builtin(__builtin_amdgcn_s_wait_asynccnt)
#define USE_ASYNC_LDS 1
#else
#define USE_ASYNC_LDS 0
#endif

#if USE_ASYNC_LDS
__device__ __forceinline__ void async_copy_b128(const float* g, float* l) {
  // Low 32 bits of a generic pointer into LDS == byte offset in workgroup LDS
  // (ISA 10.2 aperture mapping).
  unsigned lds_addr = (unsigned)(unsigned long long)l;
  asm volatile("global_load_async_to_lds_b128 %0, %1, off"
               :
               : "v"(lds_addr), "v"(g)
               : "memory");
}
__device__ __forceinline__ void async_wait_all() {
  __builtin_amdgcn_s_wait_asynccnt(0);
}
#endif

// ---------------------------------------------------------------------------
// Kernel 1: Gram matrix  G = Z Z^T + eps I   (n x n, inner dim kdim)
// ---------------------------------------------------------------------------
__global__ void gram_kernel(const float* __restrict__ Z, int n, int kdim,
                            float* __restrict__ G, float eps) {
  int id = blockIdx.x * blockDim.x + threadIdx.x;
  if (id >= n * n) return;
  int i = id / n, j = id % n;
  const float4* zi = (const float4*)(Z + (size_t)i * kdim);
  const float4* zj = (const float4*)(Z + (size_t)j * kdim);
  float acc = 0.f;
  int k4 = kdim >> 2;
  for (int t = 0; t < k4; ++t) {
    float4 a = zi[t], b = zj[t];
    acc += a.x * b.x + a.y * b.y + a.z * b.z + a.w * b.w;
  }
  if (i == j) acc += eps;
  G[id] = acc;
}

// ---------------------------------------------------------------------------
// Kernel 2: derive RHS = [Y | -H12-U] (256x320), D11 (64x64), C1 (64x256)
// ---------------------------------------------------------------------------
__global__ void derive_kernel(const float* __restrict__ H,
                              const float* __restrict__ Y1,
                              const float* __restrict__ U,
                              float* __restrict__ RHS,
                              float* __restrict__ D11,
                              float* __restrict__ C1) {
  int id = blockIdx.x * blockDim.x + threadIdx.x;
  if (id < kNX * kNH) {
    int i = id / kNH, j = id % kNH;
    float v;
    if (j < kNX) {
      v = -0.5f * (H[i * kNH + j] + Y1[i * kNX + j] - Y1[j * kNX + i]);
    } else {
      int jj = j - kNX;
      v = -H[i * kNH + kNX + jj] - U[i * kNQ + jj];
    }
    RHS[id] = v;
  } else if (id < kNX * kNH + kNQ * kNQ) {
    int idx = id - kNX * kNH;
    int i = idx >> 6, j = idx & 63;
    float lam = 0.5f * H[(kNX + i) * kNH + kNX + i];
    D11[idx] = (j < i) ? (-H[(kNX + i) * kNH + kNX + j] / lam) : 0.f;
  } else if (id < kNX * kNH + kNQ * kNQ + kNQ * kNX) {
    int idx = id - kNX * kNH - kNQ * kNQ;
    int i = idx >> 8, j = idx & 255;
    float lam = 0.5f * H[(kNX + i) * kNH + kNX + i];
    C1[idx] = U[j * kNQ + i] / lam;
  }
}

// ---------------------------------------------------------------------------
// Kernel 3: in-place Gauss-Jordan inversion of SPD P (256x256) in LDS.
// One workgroup, 1024 threads, 258 KB LDS (fits the 320 KB WGP LDS).
// ---------------------------------------------------------------------------
__global__ void __launch_bounds__(1024) invert_kernel(float* __restrict__ P) {
  extern __shared__ float smem[];
  float* Ps  = smem;                 // 256*256
  float* col = smem + 65536;         // 256
  float* row = smem + 65536 + 256;   // 256
  int tid = threadIdx.x;

  for (int e = 0; e < 16; ++e) {
    int idx = (e * 1024 + tid) * 4;
    *(float4*)(Ps + idx) = *(const float4*)(P + idx);
  }
  __syncthreads();

  for (int k = 0; k < 256; ++k) {
    if (tid < 256) {
      col[tid] = Ps[tid * 256 + k];
      row[tid] = Ps[k * 256 + tid];
    }
    __syncthreads();
    float ipiv = 1.0f / row[k];
    for (int e = 0; e < 64; ++e) {
      int idx = e * 1024 + tid;
      int i = idx >> 8, j = idx & 255;
      float v;
      if (j == k)       v = (i == k) ? ipiv : (-col[i] * ipiv);
      else if (i == k)  v = row[j] * ipiv;
      else              v = Ps[idx] - col[i] * (ipiv * row[j]);
      Ps[idx] = v;
    }
    __syncthreads();
  }

  for (int e = 0; e < 16; ++e) {
    int idx = (e * 1024 + tid) * 4;
    *(float4*)(P + idx) = *(const float4*)(Ps + idx);
  }
}

// ---------------------------------------------------------------------------
// Kernel 4: [A | B1] = Pinv @ RHS  (256 x 320) with WMMA f32 16x16x4.
// One wave per 16x16 output tile; 320 tiles -> 40 blocks of 8 waves.
// ---------------------------------------------------------------------------
__global__ void solve_gemm_kernel(const float* __restrict__ Pinv,
                                  const float* __restrict__ RHS,
                                  float* __restrict__ A,
                                  float* __restrict__ B1) {
  int lane = threadIdx.x & 31;
  int wave = threadIdx.x >> 5;
  int gw = blockIdx.x * 8 + wave;       // 0..319
  int m0 = (gw / 20) * 16;
  int n0 = (gw % 20) * 16;
  int lm = lane & 15;
  int hi = lane >> 4;
  int lk = hi * 2;

  v8f c = {};
#pragma unroll 4
  for (int k = 0; k < 256; k += 4) {
    v2f a = *(const v2f*)(Pinv + (m0 + lm) * 256 + k + lk);
    v2f b;
    b[0] = RHS[(k + lk) * kNH + n0 + lm];
    b[1] = RHS[(k + lk + 1) * kNH + n0 + lm];
    c = WMMA_F32(a, b, c);
  }
  int colg = n0 + lm;
  for (int r = 0; r < 8; ++r) {
    int rowi = m0 + r + 8 * hi;
    if (colg < kNX) A[rowi * kNX + colg] = c[r];
    else            B1[rowi * kNQ + (colg - kNX)] = c[r];
  }
}

// ---------------------------------------------------------------------------
// Kernel 5: fused main kernel. 128 batch rows per workgroup, 256 threads.
// Each wave owns a 16-row strip (m0 = wave*16) and runs 4 accumulators wide
// so every A-fragment LDS read feeds 4 independent WMMA chains.
// ---------------------------------------------------------------------------
static constexpr int ROWS = 128;
static constexpr int XS_LD  = 260;  // % 64 == 4 -> conflict-free A-frags
static constexpr int US_LD  = 12;
static constexpr int BS_LD  = 68;
static constexpr int WS_LD  = 68;
static constexpr int D11_LD = 68;

__global__ void __launch_bounds__(256) ren_main_kernel(
    const float* __restrict__ x, const float* __restrict__ u,
    const float* __restrict__ C1, const float* __restrict__ D12,
    const float* __restrict__ bv, const float* __restrict__ A,
    const float* __restrict__ B1, const float* __restrict__ B2,
    const float* __restrict__ D11, float* __restrict__ out) {
  extern __shared__ float smem[];
  float* xs   = smem;                    // ROWS*XS_LD
  float* us   = xs + ROWS * XS_LD;       // ROWS*US_LD
  float* d11s = us + ROWS * US_LD;       // 64*D11_LD
  float* bs   = d11s + 64 * D11_LD;      // ROWS*BS_LD
  float* wsm  = bs + ROWS * BS_LD;       // ROWS*WS_LD

  int tid  = threadIdx.x;
  int lane = tid & 31, wave = tid >> 5;
  int lm = lane & 15, hi = lane >> 4, lk = hi * 2;
  int row0 = blockIdx.x * ROWS;
  int m0 = wave * 16;

  // Warm A (256 KB) into L2 while we stage the x tile.
  __builtin_prefetch(A + ((size_t)tid << 8), 0, 3);

  // ---- stage x tile (128x256), u tile (128x8), D11 (64x64) into LDS ----
#if USE_ASYNC_LDS
  for (int e = 0; e < 32; ++e) {
    int lin = (e * 256 + tid) * 4;
    int r = lin >> 8, cc = lin & 255;
    async_copy_b128(x + (size_t)(row0 + r) * kNX + cc, xs + r * XS_LD + cc);
  }
  {
    int lin = tid * 4;
    int r = lin >> 3, cc = lin & 7;
    async_copy_b128(u + (size_t)(row0 + r) * 8 + cc, us + r * US_LD + cc);
  }
  for (int e = 0; e < 4; ++e) {
    int lin = (e * 256 + tid) * 4;
    int r = lin >> 6, cc = lin & 63;
    async_copy_b128(D11 + lin, d11s + r * D11_LD + cc);
  }
  async_wait_all();
#else
  for (int e = 0; e < 32; ++e) {
    int lin = (e * 256 + tid) * 4;
    int r = lin >> 8, cc = lin & 255;
    *(float4*)(xs + r * XS_LD + cc) =
        *(const float4*)(x + (size_t)(row0 + r) * kNX + cc);
  }
  {
    int lin = tid * 4;
    int r = lin >> 3, cc = lin & 7;
    *(float4*)(us + r * US_LD + cc) =
        *(const float4*)(u + (size_t)(row0 + r) * 8 + cc);
  }
  for (int e = 0; e < 4; ++e) {
    int lin = (e * 256 + tid) * 4;
    int r = lin >> 6, cc = lin & 63;
    *(float4*)(d11s + r * D11_LD + cc) = *(const float4*)(D11 + lin);
  }
#endif
  __syncthreads();

  // ---- base = x@C1^T + u@D12^T + bv  (128x64), 4 n-tiles per wave ----
  {
    v8f c0 = {}, c1 = {}, c2 = {}, c3 = {};
#pragma unroll 2
    for (int k = 0; k < 256; k += 4) {
      v2f a  = *(const v2f*)(xs + (m0 + lm) * XS_LD + k + lk);
      v2f b0 = *(const v2f*)(C1 + (lm) * kNX + k + lk);
      v2f b1 = *(const v2f*)(C1 + (16 + lm) * kNX + k + lk);
      v2f b2 = *(const v2f*)(C1 + (32 + lm) * kNX + k + lk);
      v2f b3 = *(const v2f*)(C1 + (48 + lm) * kNX + k + lk);
      c0 = WMMA_F32(a, b0, c0);
      c1 = WMMA_F32(a, b1, c1);
      c2 = WMMA_F32(a, b2, c2);
      c3 = WMMA_F32(a, b3, c3);
    }
#pragma unroll
    for (int k = 0; k < 8; k += 4) {
      v2f a  = *(const v2f*)(us + (m0 + lm) * US_LD + k + lk);
      v2f b0 = *(const v2f*)(D12 + (lm) * 8 + k + lk);
      v2f b1 = *(const v2f*)(D12 + (16 + lm) * 8 + k + lk);
      v2f b2 = *(const v2f*)(D12 + (32 + lm) * 8 + k + lk);
      v2f b3 = *(const v2f*)(D12 + (48 + lm) * 8 + k + lk);
      c0 = WMMA_F32(a, b0, c0);
      c1 = WMMA_F32(a, b1, c1);
      c2 = WMMA_F32(a, b2, c2);
      c3 = WMMA_F32(a, b3, c3);
    }
    float bv0 = bv[lm], bv1 = bv[16 + lm], bv2 = bv[32 + lm], bv3 = bv[48 + lm];
    for (int r = 0; r < 8; ++r) {
      int rr = (m0 + r + 8 * hi) * BS_LD;
      bs[rr + lm]      = c0[r] + bv0;
      bs[rr + 16 + lm] = c1[r] + bv1;
      bs[rr + 32 + lm] = c2[r] + bv2;
      bs[rr + 48 + lm] = c3[r] + bv3;
    }
  }
  __syncthreads();

  // ---- w recurrence: w[i] = relu(base[i] + sum_{j<i} w[j]*D11[i][j]) ----
  if (tid < ROWS) {
    for (int i = 0; i < 64; ++i) {
      float acc = bs[tid * BS_LD + i];
      for (int j = 0; j < i; ++j)
        acc += wsm[tid * WS_LD + j] * d11s[i * D11_LD + j];
      wsm[tid * WS_LD + i] = acc > 0.f ? acc : 0.f;
    }
  }
  __syncthreads();

  // ---- x_dot = x@A^T + w@B1^T + u@B2^T  (128x256), 4 n-tiles at a time ----
  for (int ng = 0; ng < 4; ++ng) {
    int n0 = ng * 64;
    v8f c0 = {}, c1 = {}, c2 = {}, c3 = {};
#pragma unroll 2
    for (int k = 0; k < 256; k += 4) {
      v2f a  = *(const v2f*)(xs + (m0 + lm) * XS_LD + k + lk);
      v2f b0 = *(const v2f*)(A + (n0 + lm) * kNX + k + lk);
      v2f b1 = *(const v2f*)(A + (n0 + 16 + lm) * kNX + k + lk);
      v2f b2 = *(const v2f*)(A + (n0 + 32 + lm) * kNX + k + lk);
      v2f b3 = *(const v2f*)(A + (n0 + 48 + lm) * kNX + k + lk);
      c0 = WMMA_F32(a, b0, c0);
      c1 = WMMA_F32(a, b1, c1);
      c2 = WMMA_F32(a, b2, c2);
      c3 = WMMA_F32(a, b3, c3);
    }
#pragma unroll 2
    for (int k = 0; k < 64; k += 4) {
      v2f a  = *(const v2f*)(wsm + (m0 + lm) * WS_LD + k + lk);
      v2f b0 = *(const v2f*)(B1 + (n0 + lm) * kNQ + k + lk);
      v2f b1 = *(const v2f*)(B1 + (n0 + 16 + lm) * kNQ + k + lk);
      v2f b2 = *(const v2f*)(B1 + (n0 + 32 + lm) * kNQ + k + lk);
      v2f b3 = *(const v2f*)(B1 + (n0 + 48 + lm) * kNQ + k + lk);
      c0 = WMMA_F32(a, b0, c0);
      c1 = WMMA_F32(a, b1, c1);
      c2 = WMMA_F32(a, b2, c2);
      c3 = WMMA_F32(a, b3, c3);
    }
#pragma unroll
    for (int k = 0; k < 8; k += 4) {
      v2f a  = *(const v2f*)(us + (m0 + lm) * US_LD + k + lk);
      v2f b0 = *(const v2f*)(B2 + (n0 + lm) * 8 + k + lk);
      v2f b1 = *(const v2f*)(B2 + (n0 + 16 + lm) * 8 + k + lk);
      v2f b2 = *(const v2f*)(B2 + (n0 + 32 + lm) * 8 + k + lk);
      v2f b3 = *(const v2f*)(B2 + (n0 + 48 + lm) * 8 + k + lk);
      c0 = WMMA_F32(a, b0, c0);
      c1 = WMMA_F32(a, b1, c1);
      c2 = WMMA_F32(a, b2, c2);
      c3 = WMMA_F32(a, b3, c3);
    }
    for (int r = 0; r < 8; ++r) {
      size_t rowoff = (size_t)(row0 + m0 + r + 8 * hi) * kNX;
      out[rowoff + n0 + lm]      = c0[r];
      out[rowoff + n0 + 16 + lm] = c1[r];
      out[rowoff + n0 + 32 + lm] = c2[r];
      out[rowoff + n0 + 48 + lm] = c3[r];
    }
  }
}

// ---------------------------------------------------------------------------
extern "C" void kernel_launch(void* const* d_in, const int* in_sizes, int n_in,
                              void* d_out, int out_size, void* d_ws, size_t ws_size,
                              hipStream_t stream) {
  (void)in_sizes; (void)n_in; (void)out_size; (void)ws_size;
  const float* x   = (const float*)d_in[0];
  const float* u   = (const float*)d_in[1];
  const float* X   = (const float*)d_in[2];
  const float* U   = (const float*)d_in[3];
  const float* Y1  = (const float*)d_in[4];
  const float* XP  = (const float*)d_in[5];
  const float* B2  = (const float*)d_in[6];
  const float* D12 = (const float*)d_in[7];
  const float* bv  = (const float*)d_in[8];
  float* out = (float*)d_out;

  float* ws  = (float*)d_ws;
  float* H   = ws;               // 320*320 = 102400
  float* P   = H + 102400;       // 256*256 = 65536  (becomes Pinv in place)
  float* RHS = P + 65536;        // 256*320 = 81920
  float* A   = RHS + 81920;      // 256*256 = 65536
  float* B1  = A + 65536;        // 256*64  = 16384
  float* C1  = B1 + 16384;       // 64*256  = 16384
  float* D11 = C1 + 16384;       // 64*64   = 4096

  gram_kernel<<<(320 * 320 + 255) / 256, 256, 0, stream>>>(X, 320, 320, H, 0.01f);
  gram_kernel<<<(256 * 256 + 255) / 256, 256, 0, stream>>>(XP, 256, 256, P, 0.01f);
  derive_kernel<<<(102400 + 255) / 256, 256, 0, stream>>>(H, Y1, U, RHS, D11, C1);
  invert_kernel<<<1, 1024, (65536 + 512) * sizeof(float), stream>>>(P);
  solve_gemm_kernel<<<40, 256, 0, stream>>>(P, RHS, A, B1);

  size_t smem = (size_t)(ROWS * XS_LD + ROWS * US_LD + 64 * D11_LD +
                         ROWS * BS_LD + ROWS * WS_LD) * sizeof(float);
  ren_main_kernel<<<65536 / ROWS, 256, smem, stream>>>(
      x, u, C1, D12, bv, A, B1, B2, D11, out);
}